// BWItnBlock_47064251630242
// MI455X (gfx1250) — compile-verified
//
#include <hip/hip_runtime.h>

#define NIMG 64
#define CCH  128
#define HH   56
#define WW   56
#define HWSZ (HH * WW)           // 3136 (multiple of 32 -> tiles never cross images)
#define MTOT (NIMG * HWSZ)       // 200704
#define TITER 10
#define EPSV 1e-5f

typedef __attribute__((ext_vector_type(2))) float v2f;
typedef __attribute__((ext_vector_type(8))) float v8f;

__device__ __forceinline__ v8f wmma_f32_4(v2f a, v2f b, v8f c) {
  // D(16x16,f32) = A(16x4,f32) * B(4x16,f32) + C
  return __builtin_amdgcn_wmma_f32_16x16x4_f32(false, a, false, b, (short)0, c,
                                               false, false);
}

// ---- CDNA5 async global->LDS DMA (ASYNCcnt-tracked), inline asm ----
__device__ __forceinline__ void async_ld16(const float* lds_ptr, const float* g) {
  // low 32 bits of a generic shared pointer == LDS byte offset (aperture decode)
  unsigned loff = (unsigned)(uintptr_t)lds_ptr;
  asm volatile("global_load_async_to_lds_b128 %0, %1, off"
               :
               : "v"(loff), "v"(g)
               : "memory");
}
__device__ __forceinline__ void wait_async0() {
  asm volatile("s_wait_asynccnt 0x0" ::: "memory");
}

// ---------------- kernel 0: zero the Gram accumulator ----------------
__global__ void zero_kernel(float* Gacc) {
  int i = blockIdx.x * blockDim.x + threadIdx.x;
  if (i < CCH * CCH) Gacc[i] = 0.0f;
}

// ---------------- kernel 1: per-channel mean ----------------
__global__ void mean_kernel(const float* __restrict__ X, float* __restrict__ meanb) {
  int c = blockIdx.x;          // one block per channel
  int t = threadIdx.x;         // 256 threads
  float s = 0.0f;
  for (int n = 0; n < NIMG; ++n) {
    const float* p = X + (size_t)n * CCH * HWSZ + (size_t)c * HWSZ;
    for (int i = t; i < HWSZ; i += 256) s += p[i];
  }
  __shared__ float red[256];
  red[t] = s;
  __syncthreads();
  for (int off = 128; off > 0; off >>= 1) {
    if (t < off) red[t] += red[t + off];
    __syncthreads();
  }
  if (t == 0) meanb[c] = red[0] / (float)MTOT;
}

// ------- kernel 2: raw Gram matrix G = sum_m x x^T (WMMA + async double-buffer) -------
#define MCHUNK 1024
#define LPAD 36   // 36-float row stride: 16B-aligned, bank-conflict-free fragments
__global__ void cov_kernel(const float* __restrict__ X, float* __restrict__ Gacc) {
  __shared__ float xt[2][CCH][LPAD];   // two 128x32 tiles (double buffered)
  int t = threadIdx.x;                 // 256 threads = 8 waves (wave32)
  int wv = t >> 5;
  int l  = t & 31;
  int r  = l & 15;                     // A row / B col within tile
  int half = l >> 4;                   // 0: K=0..1, 1: K=2..3
  int cload = t & 127;                 // channel this thread stages
  int part  = t >> 7;                  // 0/1 -> halves of the 32-sample row

  v8f acc[8] = {};                     // this wave's 16x128 strip: 8 16x16 tiles
  int m_start = blockIdx.x * MCHUNK;
  const int NT = MCHUNK / 32;

  // prologue: stage tile 0 into buffer 0
  {
    int m0 = m_start;
    int nimg = m0 / HWSZ;
    int hw0 = m0 - nimg * HWSZ;
    const float* gp =
        X + (size_t)nimg * CCH * HWSZ + (size_t)cload * HWSZ + hw0 + part * 16;
    const float* lp = &xt[0][cload][part * 16];
#pragma unroll
    for (int q = 0; q < 4; ++q) async_ld16(lp + 4 * q, gp + 4 * q);
  }

  for (int tile = 0; tile < NT; ++tile) {
    wait_async0();        // my async writes to current buffer are in LDS
    __syncthreads();      // everyone's are; previous compute buffer is free
    if (tile + 1 < NT) {  // kick off next tile's DMA into the other buffer
      int m0 = m_start + (tile + 1) * 32;
      int nimg = m0 / HWSZ;
      int hw0 = m0 - nimg * HWSZ;
      const float* gp =
          X + (size_t)nimg * CCH * HWSZ + (size_t)cload * HWSZ + hw0 + part * 16;
      const float* lp = &xt[(tile + 1) & 1][cload][part * 16];
#pragma unroll
      for (int q = 0; q < 4; ++q) async_ld16(lp + 4 * q, gp + 4 * q);
    }
    const float(*xb)[LPAD] = xt[tile & 1];
#pragma unroll
    for (int kk = 0; kk < 8; ++kk) {
      int k = kk * 4 + 2 * half;
      v2f a;
      a.x = xb[16 * wv + r][k];
      a.y = xb[16 * wv + r][k + 1];
#pragma unroll
      for (int j = 0; j < 8; ++j) {
        v2f b;                          // B = x^T, so B[k][c] = xt[c][k]
        b.x = xb[16 * j + r][k];
        b.y = xb[16 * j + r][k + 1];
        acc[j] = wmma_f32_4(a, b, acc[j]);
      }
    }
    __syncthreads();
  }
  // accumulate partial Gram matrix
#pragma unroll
  for (int j = 0; j < 8; ++j)
#pragma unroll
    for (int rr = 0; rr < 8; ++rr) {
      int row = rr + 8 * half;
      atomicAdd(&Gacc[(16 * wv + row) * CCH + 16 * j + r], acc[j][rr]);
    }
}

// ---------------- 128x128 matmul strip helper (single block, 8 waves) ----------------
__device__ __forceinline__ void mm128(float* __restrict__ D,
                                      const float* __restrict__ A,
                                      const float* __restrict__ B, int t) {
  int wv = t >> 5;
  int l = t & 31;
  int r = l & 15;
  int half = l >> 4;
  v8f acc[8] = {};
  for (int kk = 0; kk < 32; ++kk) {
    int k = kk * 4 + 2 * half;
    const float* ap = A + (16 * wv + r) * CCH + k;
    v2f a;
    a.x = ap[0];
    a.y = ap[1];
#pragma unroll
    for (int j = 0; j < 8; ++j) {
      v2f b;
      b.x = B[k * CCH + 16 * j + r];
      b.y = B[(k + 1) * CCH + 16 * j + r];
      acc[j] = wmma_f32_4(a, b, acc[j]);
    }
  }
#pragma unroll
  for (int j = 0; j < 8; ++j)
#pragma unroll
    for (int rr = 0; rr < 8; ++rr) {
      int row = rr + 8 * half;
      D[(16 * wv + row) * CCH + 16 * j + r] = acc[j][rr];
    }
  __threadfence();
  __syncthreads();
}

// ------- kernel 3: Sigma from Gram + mean, Newton-Schulz -> whitening matrix -------
__global__ void ns_kernel(const float* __restrict__ Gacc,
                          const float* __restrict__ meanb, float* __restrict__ Sbuf,
                          float* __restrict__ Pbuf, float* __restrict__ T1,
                          float* __restrict__ T2, float* __restrict__ WMb) {
  int t = threadIdx.x;  // 256 threads
  __shared__ float red[256];
  __shared__ float s_rTr;
  float tv = 0.0f;
  if (t < CCH) {
    float mu = meanb[t];
    tv = Gacc[t * CCH + t] / (float)MTOT - mu * mu;  // diag of (G/m - mu mu^T)
  }
  red[t] = tv;
  __syncthreads();
  for (int off = 128; off > 0; off >>= 1) {
    if (t < off) red[t] += red[t + off];
    __syncthreads();
  }
  if (t == 0) {
    float trace = red[0] + (float)CCH * EPSV;
    s_rTr = 1.0f / trace;
  }
  __syncthreads();
  float rTr = s_rTr;

  // Sigma_N = (G/m - mu mu^T + eps I) * rTr ; P = I
  for (int i = t; i < CCH * CCH; i += 256) {
    int rr = i >> 7, cc = i & 127;
    float sig = Gacc[i] / (float)MTOT - meanb[rr] * meanb[cc] +
                ((rr == cc) ? EPSV : 0.0f);
    Sbuf[i] = sig * rTr;
    Pbuf[i] = (rr == cc) ? 1.0f : 0.0f;
  }
  __threadfence();
  __syncthreads();

  for (int it = 0; it < TITER; ++it) {
    mm128(T1, Pbuf, Pbuf, t);   // P*P
    mm128(T2, T1, Pbuf, t);     // P*P*P
    mm128(T1, T2, Sbuf, t);     // P*P*P*Sigma_N
    for (int i = t; i < CCH * CCH; i += 256)
      Pbuf[i] = 1.5f * Pbuf[i] - 0.5f * T1[i];
    __threadfence();
    __syncthreads();
  }
  float srTr = sqrtf(rTr);
  for (int i = t; i < CCH * CCH; i += 256) WMb[i] = Pbuf[i] * srTr;
}

// ---------------- kernel 4: whitening GEMM + beta (WMMA) ----------------
__global__ void whiten_kernel(const float* __restrict__ X,
                              const float* __restrict__ wmg,
                              const float* __restrict__ meanb,
                              const float* __restrict__ beta,
                              float* __restrict__ out) {
  __shared__ float wls[CCH * CCH];  // 64 KB: whitening matrix staged in LDS
  int t = threadIdx.x;              // 256 threads = 8 waves
  for (int i = t; i < CCH * CCH; i += 256) wls[i] = wmg[i];
  __syncthreads();

  int wv = t >> 5;
  int l = t & 31;
  int r = l & 15;
  int half = l >> 4;
  int mcol = blockIdx.x * 128 + wv * 16 + r;  // this lane's m column
  int nimg = mcol / HWSZ;
  int hw = mcol - nimg * HWSZ;
  size_t base = (size_t)nimg * CCH * HWSZ + hw;

  v8f acc[8] = {};
  for (int kk = 0; kk < 32; ++kk) {
    int k = kk * 4 + 2 * half;
    v2f b;  // B[k][n] = xc[channel k][m column n] — coalesced along lanes
    b.x = X[base + (size_t)k * HWSZ] - meanb[k];
    b.y = X[base + (size_t)(k + 1) * HWSZ] - meanb[k + 1];
#pragma unroll
    for (int i = 0; i < 8; ++i) {
      v2f a;  // A[r][k] = wm[16*i + r][k]
      a.x = wls[(16 * i + r) * CCH + k];
      a.y = wls[(16 * i + r) * CCH + k + 1];
      acc[i] = wmma_f32_4(a, b, acc[i]);
    }
  }
#pragma unroll
  for (int i = 0; i < 8; ++i)
#pragma unroll
    for (int rr = 0; rr < 8; ++rr) {
      int c = 16 * i + rr + 8 * half;
      out[base + (size_t)c * HWSZ] = acc[i][rr] + beta[c];
    }
}

extern "C" void kernel_launch(void* const* d_in, const int* in_sizes, int n_in,
                              void* d_out, int out_size, void* d_ws, size_t ws_size,
                              hipStream_t stream) {
  const float* X = (const float*)d_in[0];
  const float* beta = (const float*)d_in[1];
  // d_in[2]=running_mean, d_in[3]=running_cov: unused (output is X_hat only)
  float* ws = (float*)d_ws;
  float* meanb = ws;                 // 128
  float* Gacc = ws + 128;            // 16384 (raw Gram accumulator)
  float* Sbuf = Gacc + CCH * CCH;    // 16384
  float* Pbuf = Sbuf + CCH * CCH;    // 16384
  float* T1 = Pbuf + CCH * CCH;      // 16384
  float* T2 = T1 + CCH * CCH;        // 16384
  float* WMb = T2 + CCH * CCH;       // 16384
  float* out = (float*)d_out;

  zero_kernel<<<(CCH * CCH + 255) / 256, 256, 0, stream>>>(Gacc);
  mean_kernel<<<CCH, 256, 0, stream>>>(X, meanb);
  cov_kernel<<<MTOT / MCHUNK, 256, 0, stream>>>(X, Gacc);
  ns_kernel<<<1, 256, 0, stream>>>(Gacc, meanb, Sbuf, Pbuf, T1, T2, WMb);
  whiten_kernel<<<MTOT / 128, 256, 0, stream>>>(X, WMb, meanb, beta, out);
}